// InvPolarTransform_69947837383180
// MI455X (gfx1250) — compile-verified
//
#include <hip/hip_runtime.h>
#include <math.h>

#define OUT_H    512
#define OUT_W    512
#define BATCH    8
#define N_THETA  720
#define N_R      362
#define CHANNELS 32

typedef float vf4 __attribute__((ext_vector_type(4)));

__device__ __forceinline__ vf4 lerp4(vf4 a, vf4 b, float t) {
    vf4 r;
    r.x = fmaf(t, b.x - a.x, a.x);
    r.y = fmaf(t, b.y - a.y, a.y);
    r.z = fmaf(t, b.z - a.z, a.z);
    r.w = fmaf(t, b.w - a.w, a.w);
    return r;
}

__global__ __launch_bounds__(256) void InvPolarTransform_kernel(
    const float* __restrict__ img, float* __restrict__ out)
{
    __shared__ int   s_y0[32];
    __shared__ int   s_x0[32];
    __shared__ float s_ay[32];
    __shared__ float s_ax[32];

    const int b       = blockIdx.y;
    const int pixBase = blockIdx.x * 32;
    const int t       = threadIdx.x;

    // First wave: per-pixel inverse-polar coordinates (once per pixel, not per lane).
    if (t < 32) {
        const int p = pixBase + t;
        const int h = p >> 9;      // p / 512
        const int w = p & 511;     // p % 512
        const float xr = (float)h - 256.0f;   // row offset  (x in reference)
        const float yc = (float)w - 256.0f;   // col offset  (y in reference)

        const float TWO_PI  = 6.28318530717958647692f;
        const float K_THETA = (float)N_THETA / TWO_PI;            // 720 / 2pi
        const float K_R     = (float)N_R / 362.03867196751236f;   // 362 / (256*sqrt(2))

        float ang = atan2f(yc, xr);
        if (ang < 0.0f) ang += TWO_PI;        // mod(atan2, 2pi)

        float c1 = K_THETA * ang;
        float c2 = K_R * sqrtf(fmaf(xr, xr, yc * yc));
        c1 = fminf(fmaxf(c1, 0.0f), (float)(N_THETA - 1));
        c2 = fminf(fmaxf(c2, 0.0f), (float)(N_R - 1));

        int y0 = (int)floorf(c1);
        y0 = min(max(y0, 0), N_THETA - 2);
        int x0 = (int)floorf(c2);
        x0 = min(max(x0, 0), N_R - 2);

        s_y0[t] = y0;
        s_x0[t] = x0;
        s_ay[t] = fminf(fmaxf(c1 - (float)y0, 0.0f), 1.0f);
        s_ax[t] = fminf(fmaxf(c2 - (float)x0, 0.0f), 1.0f);
    }
    __syncthreads();

    // 8 lanes per pixel; lane handles a float4 slice of the 32 channels.
    const int group = t >> 3;    // pixel within block (0..31)
    const int lane  = t & 7;     // channel quad (0..7)

    const int   y0 = s_y0[group];
    const int   x0 = s_x0[group];
    const float ay = s_ay[group];
    const float ax = s_ax[group];

    // ((b*N_THETA + y0)*N_R + x0)*C + lane*4
    const size_t base00 =
        ((((size_t)b * N_THETA + (size_t)y0) * N_R + (size_t)x0) * CHANNELS)
        + (size_t)(lane * 4);
    const size_t rowStride = (size_t)N_R * CHANNELS;

    // Four fully-coalesced 128B gathers (8 lanes x float4 each).
    const vf4 g00 = *(const vf4*)(img + base00);
    const vf4 g01 = *(const vf4*)(img + base00 + CHANNELS);
    const vf4 g10 = *(const vf4*)(img + base00 + rowStride);
    const vf4 g11 = *(const vf4*)(img + base00 + rowStride + CHANNELS);

    const vf4 top = lerp4(g00, g01, ax);
    const vf4 bot = lerp4(g10, g11, ax);
    const vf4 res = lerp4(top, bot, ay);

    // Streaming write-once output: non-temporal so it does not evict the
    // per-batch input image (33 MB) resident in the 192 MB L2.
    const size_t p = (size_t)pixBase + (size_t)group;
    float* outp = out + ((size_t)b * (size_t)(OUT_H * OUT_W) + p) * CHANNELS
                      + (size_t)(lane * 4);
    __builtin_nontemporal_store(res, (vf4*)outp);
}

extern "C" void kernel_launch(void* const* d_in, const int* in_sizes, int n_in,
                              void* d_out, int out_size, void* d_ws, size_t ws_size,
                              hipStream_t stream) {
    (void)in_sizes; (void)n_in; (void)out_size; (void)d_ws; (void)ws_size;
    const float* img = (const float*)d_in[0];
    float* out = (float*)d_out;

    // 512*512 pixels / 32 pixels-per-block = 8192 blocks; batch in grid.y so
    // co-resident blocks share one batch's input working set in L2.
    dim3 grid((OUT_H * OUT_W) / 32, BATCH, 1);
    dim3 block(256, 1, 1);
    InvPolarTransform_kernel<<<grid, block, 0, stream>>>(img, out);
}